// HSumGraph_28235115003912
// MI455X (gfx1250) — compile-verified
//
#include <hip/hip_runtime.h>

#define Hd   300
#define HP   304            // H padded to multiple of 16
#define Bb   64
#define Nn   512
#define KQ   75             // K blocks of 4 covering Hd=300
#define CT   19             // HP/16 column tiles
#define EPSV 1e-6f

typedef __attribute__((ext_vector_type(2))) float v2f;
typedef __attribute__((ext_vector_type(8))) float v8f;

static __device__ __forceinline__ v8f wmma_f32(v2f a, v2f b, v8f c) {
    // V_WMMA_F32_16X16X4_F32: D = A(16x4) * B(4x16) + C(16x16), fp32
    return __builtin_amdgcn_wmma_f32_16x16x4_f32(
        /*neg_a=*/false, a, /*neg_b=*/false, b,
        /*c_mod=*/(short)0, c, /*reuse_a=*/false, /*reuse_b=*/false);
}

// B-operand lane layout for one 16x16-col x 4-K tile block:
//   lane = ((k & 2) ? 16 : 0) + col%16 ; pair element p = k & 1
// Packed index: ((ct*KQ + kq)*32 + lane)*2 + p

// ---------------------------------------------------------------------------
// P0: pack Wtp/Wtc/Wbil (300x300 row-major) into WMMA-B layout, zero-padded.
// Done once; removes all guards/strided loads from the GEMM hot loops.
// ---------------------------------------------------------------------------
__global__ void k_packw(const float* __restrict__ Wtp, const float* __restrict__ Wtc,
                        const float* __restrict__ Wbil, float* __restrict__ out)
{
    const int NPW = CT * KQ * 32;                 // 45600 lanes per weight
    int idx = blockIdx.x * 256 + threadIdx.x;
    if (idx >= 3 * NPW) return;
    const int which = idx / NPW;
    const int r     = idx - which * NPW;
    const int ct    = r / (KQ * 32);
    const int r2    = r - ct * (KQ * 32);
    const int kq    = r2 >> 5;
    const int lane  = r2 & 31;
    const float* Wm = (which == 0) ? Wtp : (which == 1) ? Wtc : Wbil;
    const int col = ct * 16 + (lane & 15);
    const int k0  = kq * 4 + 2 * (lane >> 4);     // k0+1 <= 299 always
    float v0 = 0.f, v1 = 0.f;
    if (col < Hd) {
        v0 = Wm[(size_t)k0 * Hd + col];
        v1 = Wm[(size_t)(k0 + 1) * Hd + col];
    }
    float* o = out + (size_t)idx * 2;
    o[0] = v0; o[1] = v1;
}

// ---------------------------------------------------------------------------
// K0: tp = relu(X @ Wtp + b), written row-major padded (A-operand downstream).
//     tc = relu(X @ Wtc + b), written DIRECTLY in WMMA-B packed layout
//     (tc is only ever consumed as the B operand of the scores GEMM).
// Inner loop: 1 coalesced b64 A load + 1 coalesced b64 B load + 1 WMMA.
// ---------------------------------------------------------------------------
__global__ void k_proj(const float* __restrict__ X,
                       const float* __restrict__ wpTP, const float* __restrict__ btp,
                       const float* __restrict__ wpTC, const float* __restrict__ btc,
                       float* __restrict__ tp, float* __restrict__ tcp)
{
    const int lane = threadIdx.x & 31;
    const int half = lane >> 4, l16 = lane & 15;
    const int MT = (Bb * Nn) / 16;                // 2048 row tiles
    int tile  = blockIdx.x * 8 + (threadIdx.x >> 5);
    int which = tile / (MT * CT);
    int rem   = tile - which * (MT * CT);
    int mt = rem / CT, ct = rem % CT;

    const float* wp = which ? wpTC : wpTP;
    const float* bm = which ? btc  : btp;

    const float* xr = X + (size_t)(mt * 16 + l16) * Hd;
    const float* bp = wp + ((size_t)(ct * KQ) * 32 + lane) * 2;

    v8f acc = {};
    for (int kq = 0; kq < KQ; ++kq) {
        const int ka = kq * 4 + 2 * half;
        v2f a; a[0] = xr[ka]; a[1] = xr[ka + 1];
        v2f b = *(const v2f*)(bp + (size_t)kq * 64);
        acc = wmma_f32(a, b, acc);
    }

    const int d = ct * 16 + l16;                  // output column / feature index
    const float bias = (d < Hd) ? bm[d] : 0.f;
    if (which == 0) {
        // tp: row-major, HP-padded (pad columns written as zero)
#pragma unroll
        for (int r = 0; r < 8; ++r) {
            const int orow = mt * 16 + r + 8 * half;
            const float v = (d < Hd) ? fmaxf(acc[r] + bias, 0.f) : 0.f;
            tp[(size_t)orow * HP + d] = v;
        }
    } else if (d < Hd) {
        // tc: scatter into packed-B layout [b][mtile][kq][lane][2]
        const int kq = d >> 2, hd = (d >> 1) & 1, p = d & 1;
#pragma unroll
        for (int r = 0; r < 8; ++r) {
            const int mg   = mt * 16 + r + 8 * half;   // global row (b*512 + m)
            const int b_   = mg >> 9;
            const int mloc = mg & 511;
            const size_t addr =
                ((((size_t)b_ * 32 + (mloc >> 4)) * KQ + kq) * 32 + (hd * 16 + (mloc & 15))) * 2 + p;
            tcp[addr] = fmaxf(acc[r] + bias, 0.f);
        }
    }
}

// ---------------------------------------------------------------------------
// K1: tpW = tp @ Wbil (packed B), row-major HP-padded output
// ---------------------------------------------------------------------------
__global__ void k_bilinear(const float* __restrict__ tp,
                           const float* __restrict__ wpBIL,
                           float* __restrict__ tpW)
{
    const int lane = threadIdx.x & 31;
    const int half = lane >> 4, l16 = lane & 15;
    int tile = blockIdx.x * 8 + (threadIdx.x >> 5);
    int mt = tile / CT, ct = tile % CT;

    const float* ar = tp + (size_t)(mt * 16 + l16) * HP;
    const float* bp = wpBIL + ((size_t)(ct * KQ) * 32 + lane) * 2;

    v8f acc = {};
    for (int kq = 0; kq < KQ; ++kq) {
        const int ka = kq * 4 + 2 * half;
        v2f a; a[0] = ar[ka]; a[1] = ar[ka + 1];
        v2f b = *(const v2f*)(bp + (size_t)kq * 64);
        acc = wmma_f32(a, b, acc);
    }
    const int d = ct * 16 + l16;
#pragma unroll
    for (int r = 0; r < 8; ++r) {
        const int orow = mt * 16 + r + 8 * half;
        tpW[(size_t)orow * HP + d] = (d < Hd) ? acc[r] : 0.f;
    }
}

// ---------------------------------------------------------------------------
// K2: scores[b,n,m] = sum_d tpW[b,n,d]*tc[b,m,d]; A = (relu*mask*mask+EPS)*(1-eye)
// B operand comes pre-packed (tcp) -> fully coalesced b64 loads.
// ---------------------------------------------------------------------------
__global__ void k_scores(const float* __restrict__ tpW,
                         const float* __restrict__ tcp,
                         const float* __restrict__ mask,
                         float* __restrict__ A)
{
    const int lane = threadIdx.x & 31;
    const int half = lane >> 4, l16 = lane & 15;
    int tile = blockIdx.x * 8 + (threadIdx.x >> 5);  // 64*32*32 tiles
    int b   = tile >> 10;
    int rem = tile & 1023;
    int nt = rem >> 5, mt = rem & 31;

    const float* ar = tpW + (size_t)(b * Nn + nt * 16 + l16) * HP;
    const float* bp = tcp + (((size_t)b * 32 + mt) * KQ * 32 + lane) * 2;

    v8f acc = {};
    for (int kq = 0; kq < KQ; ++kq) {
        const int ka = kq * 4 + 2 * half;
        v2f a; a[0] = ar[ka]; a[1] = ar[ka + 1];
        v2f b2 = *(const v2f*)(bp + (size_t)kq * 64);
        acc = wmma_f32(a, b2, acc);
    }
#pragma unroll
    for (int r = 0; r < 8; ++r) {
        const int n = nt * 16 + r + 8 * half;
        const int m = mt * 16 + l16;
        float s = fmaxf(acc[r], 0.f) * mask[b * Nn + m] * mask[b * Nn + n] + EPSV;
        if (n == m) s = 0.f;
        A[((size_t)b * Nn + n) * Nn + m] = s;
    }
}

// ---------------------------------------------------------------------------
// K3: f_i[b,n] = relu(tp[b,n,:] . wfi) * mask + EPS   (one wave per row)
// ---------------------------------------------------------------------------
__global__ void k_root(const float* __restrict__ tp, const float* __restrict__ wfi,
                       const float* __restrict__ mask, float* __restrict__ fi)
{
    const int idx  = blockIdx.x * 8 + (threadIdx.x >> 5);
    const int lane = threadIdx.x & 31;
    const float* r = tp + (size_t)idx * HP;
    float s = 0.f;
    for (int d = lane; d < Hd; d += 32) s += r[d] * wfi[d];
#pragma unroll
    for (int off = 16; off > 0; off >>= 1) s += __shfl_xor(s, off, 32);
    if (lane == 0) fi[idx] = fmaxf(s, 0.f) * mask[idx] + EPSV;
}

// ---------------------------------------------------------------------------
// K4: colsum[b,m] = sum_n A[b,n,m]   (coalesced column sums)
// ---------------------------------------------------------------------------
__global__ void k_colsum(const float* __restrict__ A, float* __restrict__ cs)
{
    const int t = blockIdx.x * 256 + threadIdx.x;
    const int b = t >> 9, m = t & 511;
    const float* Ab = A + (size_t)b * Nn * Nn;
    float s = 0.f;
    for (int n = 0; n < Nn; ++n) s += Ab[(size_t)n * Nn + m];
    cs[t] = s;
}

// ---------------------------------------------------------------------------
// K5: W = L_bar:  row0 = f_i;  else -A + diag(colsum)
// ---------------------------------------------------------------------------
__global__ void k_build(const float* __restrict__ A, const float* __restrict__ fi,
                        const float* __restrict__ cs, float* __restrict__ W)
{
    const size_t idx = (size_t)blockIdx.x * 256 + threadIdx.x;
    const int b = (int)(idx >> 18);
    const int r = (int)((idx >> 9) & 511);
    const int c = (int)(idx & 511);
    float v;
    if (r == 0) {
        v = fi[b * Nn + c];
    } else {
        v = -A[idx];
        if (r == c) v += cs[b * Nn + c];
    }
    W[idx] = v;
}

// ---------------------------------------------------------------------------
// K6: in-place Gauss-Jordan inversion of W[b] (512x512), one WG per batch.
// float4-vectorized elimination (b128 global + LDS); matrix stays in L2.
// ---------------------------------------------------------------------------
__global__ void k_invert(float* __restrict__ W)
{
    float* M = W + (size_t)blockIdx.x * Nn * Nn;
    __shared__ __align__(16) float rowk[Nn];
    __shared__ float s_pinv;
    const int t = threadIdx.x;            // 0..1023
    const int wave = t >> 5, lane = t & 31;

    for (int k = 0; k < Nn; ++k) {
        if (t == 0) s_pinv = 1.0f / M[(size_t)k * Nn + k];
        __syncthreads();
        const float pinv = s_pinv;
        if (t < Nn / 4) {
            const int j = t * 4;
            float4 mv = *(const float4*)&M[(size_t)k * Nn + j];
            float4 v;
            v.x = (j + 0 == k) ? pinv : mv.x * pinv;
            v.y = (j + 1 == k) ? pinv : mv.y * pinv;
            v.z = (j + 2 == k) ? pinv : mv.z * pinv;
            v.w = (j + 3 == k) ? pinv : mv.w * pinv;
            *(float4*)&rowk[j] = v;
            *(float4*)&M[(size_t)k * Nn + j] = v;
        }
        __syncthreads();
        for (int i = wave; i < Nn; i += 32) {
            if (i == k) continue;
            const float f   = M[(size_t)i * Nn + k];  // read before any lane writes col k
            const float nfp = -f * pinv;
            for (int jb = lane * 4; jb < Nn; jb += 128) {
                float4 mv = *(const float4*)&M[(size_t)i * Nn + jb];
                float4 rv = *(const float4*)&rowk[jb];
                float4 o;
                o.x = (jb + 0 == k) ? nfp : fmaf(-f, rv.x, mv.x);
                o.y = (jb + 1 == k) ? nfp : fmaf(-f, rv.y, mv.y);
                o.z = (jb + 2 == k) ? nfp : fmaf(-f, rv.z, mv.z);
                o.w = (jb + 3 == k) ? nfp : fmaf(-f, rv.w, mv.w);
                *(float4*)&M[(size_t)i * Nn + jb] = o;
            }
        }
        __syncthreads();
    }
}

// ---------------------------------------------------------------------------
// K7: dx[b,n,m] = (m!=0)*A*Winv[m,m] - (n!=0)*A*Winv[m,n];  d0 = f_i*Winv[:,0]
// ---------------------------------------------------------------------------
__global__ void k_out(const float* __restrict__ A, const float* __restrict__ W,
                      const float* __restrict__ fi,
                      float* __restrict__ dx, float* __restrict__ d0)
{
    const size_t idx = (size_t)blockIdx.x * 256 + threadIdx.x;
    const int b = (int)(idx >> 18);
    const int n = (int)((idx >> 9) & 511);
    const int m = (int)(idx & 511);
    const float* Wb = W + (size_t)b * Nn * Nn;
    const float a = A[idx];
    float v = 0.f;
    if (m != 0) v  = a * Wb[(size_t)m * Nn + m];
    if (n != 0) v -= a * Wb[(size_t)m * Nn + n];
    dx[idx] = v;
    if (m == 0) d0[b * Nn + n] = fi[b * Nn + n] * Wb[(size_t)n * Nn];
}

// ---------------------------------------------------------------------------
extern "C" void kernel_launch(void* const* d_in, const int* in_sizes, int n_in,
                              void* d_out, int out_size, void* d_ws, size_t ws_size,
                              hipStream_t stream)
{
    (void)in_sizes; (void)n_in; (void)out_size; (void)ws_size;
    const float* X    = (const float*)d_in[0];
    const float* msk  = (const float*)d_in[1];
    const float* Wtp  = (const float*)d_in[2];
    const float* btp  = (const float*)d_in[3];
    const float* Wtc  = (const float*)d_in[4];
    const float* btc  = (const float*)d_in[5];
    const float* Wbil = (const float*)d_in[6];
    const float* wfi  = (const float*)d_in[7];

    float* ws = (float*)d_ws;
    const size_t MP  = (size_t)Bb * Nn * HP;          // 9,961,472  (padded activation)
    const size_t NA  = (size_t)Bb * Nn * Nn;          // 16,777,216 (per-batch NxN)
    const size_t PKW = (size_t)CT * KQ * 32 * 2;      // 91,200     (one packed weight)
    const size_t PKC = (size_t)Bb * 32 * KQ * 32 * 2; // 9,830,400  (packed tc)

    float* tp   = ws;                      // [0, MP)
    float* tcp  = ws + MP;                 // packed tc
    float* tpW  = ws + MP + PKC;
    float* A    = ws + 2 * MP + PKC;
    float* fi   = A + NA;
    float* cs   = fi + Bb * Nn;
    float* wpk  = cs + Bb * Nn;            // 3 packed weights back-to-back
    float* wpTP = wpk;
    float* wpTC = wpk + PKW;
    float* wpBL = wpk + 2 * PKW;
    float* W    = ws;                      // aliases tp+tcp (dead before k_build)

    float* dx = (float*)d_out;
    float* d0 = dx + NA;

    k_packw   <<<535,   256, 0, stream>>>(Wtp, Wtc, Wbil, wpk);
    k_proj    <<<9728,  256, 0, stream>>>(X, wpTP, btp, wpTC, btc, tp, tcp);
    k_bilinear<<<4864,  256, 0, stream>>>(tp, wpBL, tpW);
    k_root    <<<4096,  256, 0, stream>>>(tp, wfi, msk, fi);
    k_scores  <<<8192,  256, 0, stream>>>(tpW, tcp, msk, A);
    k_colsum  <<<128,   256, 0, stream>>>(A, cs);
    k_build   <<<65536, 256, 0, stream>>>(A, fi, cs, W);
    k_invert  <<<64,   1024, 0, stream>>>(W);
    k_out     <<<65536, 256, 0, stream>>>(A, W, fi, dx, d0);
}